// S4Kernel_24060406792896
// MI455X (gfx1250) — compile-verified
//
#include <hip/hip_runtime.h>
#include <math.h>

typedef float v2f __attribute__((ext_vector_type(2)));
typedef float v8f __attribute__((ext_vector_type(8)));

#define L_MAX   4096
#define H_CH    256
#define NSTATE  64
#define BATCH   16
#define TWO_PI_F 6.283185307179586f

// ---------------------------------------------------------------------------
// Stage addressing: element j of group g lives at base(g) + j*stride.
//   stage 0: stride 256 (outer radix-16 pass)
//   stage 1: stride 16
//   stage 2: stride 1
// ---------------------------------------------------------------------------
__device__ __forceinline__ int stage_base(int stage, int g) {
    if (stage == 0) return g;
    if (stage == 1) return ((g >> 4) << 8) | (g & 15);
    return g << 4;
}

// B fragments of the 16x16 DFT matrix F[k][n] = exp(sgn*2*pi*i*k*n/16).
// sgn = -1 forward, +1 inverse.  Assumed 4x16 B layout for wmma_f32_16x16x4:
//   lanes 0-15 : VGPRx = B[K=0][n=lane], VGPRy = B[K=1][n=lane]
//   lanes16-31 : VGPRx = B[K=2][n],      VGPRy = B[K=3][n]
__device__ __forceinline__ void make_bfrag(int lane, float sgn,
                                           v2f br[4], v2f bi[4], v2f bin[4]) {
    const int n    = lane & 15;
    const int koff = (lane >> 4) << 1;
#pragma unroll
    for (int c = 0; c < 4; ++c) {
        int k0 = c * 4 + koff;
        int k1 = k0 + 1;
        float a0 = (TWO_PI_F / 16.0f) * (float)((k0 * n) & 15);
        float a1 = (TWO_PI_F / 16.0f) * (float)((k1 * n) & 15);
        float s0, c0, s1, c1;
        __sincosf(a0, &s0, &c0);
        __sincosf(a1, &s1, &c1);
        br[c].x = c0;        br[c].y = c1;
        bi[c].x = sgn * s0;  bi[c].y = sgn * s1;
        bin[c].x = -bi[c].x; bin[c].y = -bi[c].y;
    }
}

// One radix-16 stage over LDS, in place.  256 groups of 16 -> 16 WMMA tiles,
// 8 waves x 2 tiles.  Complex GEMM  D = Z * F  via 4 real f32 WMMAs per chunk.
// TWMODE: 0 none; 1 fwd out-twiddle W4096^(k*g); 2 fwd out-twiddle W256;
//         3 inv in-twiddle conj W256; 4 inv in-twiddle conj W4096.
template <int STAGE, int TWMODE>
__device__ __forceinline__ void fft_stage(float* zr, float* zi, int lane, int wave,
                                          const v2f br[4], const v2f bi[4],
                                          const v2f bin[4]) {
    const int stride = (STAGE == 0) ? 256 : (STAGE == 1) ? 16 : 1;
    const int koff   = (lane >> 4) << 1;
    const int r15    = lane & 15;
#pragma unroll
    for (int tt = 0; tt < 2; ++tt) {
        const int t    = wave * 2 + tt;
        const int g    = t * 16 + r15;              // A-matrix row for this lane
        const int base = stage_base(STAGE, g);
        v8f dr = {0.f, 0.f, 0.f, 0.f, 0.f, 0.f, 0.f, 0.f};
        v8f di = {0.f, 0.f, 0.f, 0.f, 0.f, 0.f, 0.f, 0.f};
#pragma unroll
        for (int c = 0; c < 4; ++c) {
            const int j0 = c * 4 + koff;
            const int j1 = j0 + 1;
            v2f ar, ai;
            ar.x = zr[base + j0 * stride];  ar.y = zr[base + j1 * stride];
            ai.x = zi[base + j0 * stride];  ai.y = zi[base + j1 * stride];
            if (TWMODE == 3 || TWMODE == 4) {       // input conj twiddle e^{+i th}
                float th0 = (TWMODE == 3)
                          ? (TWO_PI_F / 256.0f)  * (float)(j0 * (g & 15))
                          : (TWO_PI_F / 4096.0f) * (float)(j0 * g);
                float th1 = (TWMODE == 3)
                          ? (TWO_PI_F / 256.0f)  * (float)(j1 * (g & 15))
                          : (TWO_PI_F / 4096.0f) * (float)(j1 * g);
                float s0, c0, s1, c1;
                __sincosf(th0, &s0, &c0);
                __sincosf(th1, &s1, &c1);
                float xr = ar.x, xi = ai.x;
                ar.x = xr * c0 - xi * s0;  ai.x = xr * s0 + xi * c0;
                xr = ar.y; xi = ai.y;
                ar.y = xr * c1 - xi * s1;  ai.y = xr * s1 + xi * c1;
            }
            // Dr += Ar*Br + Ai*(-Bi) ;  Di += Ar*Bi + Ai*Br
            dr = __builtin_amdgcn_wmma_f32_16x16x4_f32(false, ar, false, br[c],
                                                       (short)0, dr, false, false);
            dr = __builtin_amdgcn_wmma_f32_16x16x4_f32(false, ai, false, bin[c],
                                                       (short)0, dr, false, false);
            di = __builtin_amdgcn_wmma_f32_16x16x4_f32(false, ar, false, bi[c],
                                                       (short)0, di, false, false);
            di = __builtin_amdgcn_wmma_f32_16x16x4_f32(false, ai, false, br[c],
                                                       (short)0, di, false, false);
        }
        // D layout: VGPR j -> row j (lanes 0-15) / row j+8 (lanes 16-31), col lane&15
        const int rowoff = (lane >> 4) << 3;
#pragma unroll
        for (int j = 0; j < 8; ++j) {
            const int go   = t * 16 + j + rowoff;
            const int N    = r15;
            const int addr = stage_base(STAGE, go) + N * stride;
            float orr = dr[j], oii = di[j];
            if (TWMODE == 1 || TWMODE == 2) {       // forward twiddle e^{-i th}
                float th = (TWMODE == 1)
                         ? (TWO_PI_F / 4096.0f) * (float)(N * go)
                         : (TWO_PI_F / 256.0f)  * (float)(N * (go & 15));
                float s, cc;
                __sincosf(th, &s, &cc);
                float tr = orr * cc + oii * s;
                float ti = oii * cc - orr * s;
                orr = tr; oii = ti;
            }
            zr[addr] = orr;
            zi[addr] = oii;
        }
    }
}

// ---------------------------------------------------------------------------
// Phase 1: S4 DPLR frequency kernel (Cauchy sums).  One block per channel h.
// Stores Kf[h][rev16(i)] / 4096 at ws[h*4096 + i]  (digit-reversed, prescaled)
// so the conv kernel's pointwise multiply is a linear read.
// ---------------------------------------------------------------------------
__global__ __launch_bounds__(256) void s4_kernelgen(
    const float* __restrict__ Lr_g, const float* __restrict__ Li_g,
    const float* __restrict__ Pr_g, const float* __restrict__ Pi_g,
    const float* __restrict__ Br_g, const float* __restrict__ Bi_g,
    const float* __restrict__ Cr_g, const float* __restrict__ Ci_g,
    const float* __restrict__ log_step, float2* __restrict__ KfOut) {
    __shared__ float sLr[NSTATE], sLi[NSTATE];
    __shared__ float v0r[NSTATE], v0i[NSTATE];   // conj(C)*B
    __shared__ float v1r[NSTATE], v1i[NSTATE];   // conj(C)*P
    __shared__ float v2r[NSTATE], v2i[NSTATE];   // conj(P)*B
    __shared__ float v3r[NSTATE];                // conj(P)*P  (real)
    const int h   = blockIdx.x;
    const int tid = threadIdx.x;
    if (tid < NSTATE) {
        const int o = h * NSTATE + tid;
        float lr = Lr_g[o], li = Li_g[o];
        float pr = Pr_g[o], pi = Pi_g[o];
        float brv = Br_g[o], biv = Bi_g[o];
        float cr = Cr_g[o], ci = Ci_g[o];
        sLr[tid] = lr;  sLi[tid] = li;
        v0r[tid] = cr * brv + ci * biv;  v0i[tid] = cr * biv - ci * brv;
        v1r[tid] = cr * pr + ci * pi;    v1i[tid] = cr * pi - ci * pr;
        v2r[tid] = pr * brv + pi * biv;  v2i[tid] = pr * biv - pi * brv;
        v3r[tid] = pr * pr + pi * pi;
    }
    __syncthreads();
    const float two_over_step = 2.0f * __expf(-log_step[h]);
#pragma unroll 1
    for (int it = 0; it < L_MAX / 256; ++it) {
        const int i = tid + 256 * it;                        // storage index
        const int l = ((i & 15) << 8) | (i & 0xF0) | (i >> 8); // base-16 digit rev
        float s, c;
        __sincosf((TWO_PI_F / 4096.0f) * (float)l, &s, &c);
        const float wr = c, wi = -s;                         // Omega
        // 1/(1+Omega), c = 2/(1+Omega), g = (2/step)(1-Omega)/(1+Omega)
        float opr = 1.0f + wr, opi = wi;
        float idn = 1.0f / (opr * opr + opi * opi);
        float invr = opr * idn, invi = -opi * idn;
        float c2r = 2.0f * invr, c2i = 2.0f * invi;
        float omr = 1.0f - wr, omi = -wi;
        float gr = two_over_step * (omr * invr - omi * invi);
        float gi = two_over_step * (omr * invi + omi * invr);
        float k00r = 0.f, k00i = 0.f, k01r = 0.f, k01i = 0.f;
        float k10r = 0.f, k10i = 0.f, k11r = 0.f, k11i = 0.f;
#pragma unroll 4
        for (int n = 0; n < NSTATE; ++n) {
            float dr = gr - sLr[n], di = gi - sLi[n];
            float id = 1.0f / (dr * dr + di * di);
            float rr = dr * id, ri = -di * id;
            k00r += v0r[n] * rr - v0i[n] * ri;  k00i += v0r[n] * ri + v0i[n] * rr;
            k01r += v1r[n] * rr - v1i[n] * ri;  k01i += v1r[n] * ri + v1i[n] * rr;
            k10r += v2r[n] * rr - v2i[n] * ri;  k10i += v2r[n] * ri + v2i[n] * rr;
            k11r += v3r[n] * rr;                k11i += v3r[n] * ri;
        }
        // Kf = c * (k00 - k01 * k10 / (1 + k11))
        float tr = 1.0f + k11r, ti = k11i;
        float itd = 1.0f / (tr * tr + ti * ti);
        float itr = tr * itd, iti = -ti * itd;
        float mr = k01r * k10r - k01i * k10i;
        float mi = k01r * k10i + k01i * k10r;
        float qr = mr * itr - mi * iti;
        float qi = mr * iti + mi * itr;
        float kr = k00r - qr, ki = k00i - qi;
        float Kr = c2r * kr - c2i * ki;
        float Ki = c2r * ki + c2i * kr;
        KfOut[(size_t)h * L_MAX + i] =
            make_float2(Kr * (1.0f / 4096.0f), Ki * (1.0f / 4096.0f));
    }
}

// ---------------------------------------------------------------------------
// Phase 2: FFT circular convolution.  One WG per (channel, batch-pair):
// two real rows packed as Re/Im of one complex FFT (shared Kf => linear mix).
// ---------------------------------------------------------------------------
__global__ __launch_bounds__(256) void s4_fftconv(
    const float* __restrict__ u, const float2* __restrict__ Kf,
    float* __restrict__ y) {
    __shared__ float zr[L_MAX];
    __shared__ float zi[L_MAX];
    const int h    = blockIdx.x;
    const int p    = blockIdx.y;
    const int tid  = threadIdx.x;
    const int lane = tid & 31;
    const int wave = tid >> 5;

    const float* u0 = u + ((size_t)(2 * p)     * H_CH + h) * L_MAX;
    const float* u1 = u + ((size_t)(2 * p + 1) * H_CH + h) * L_MAX;
    for (int i = tid; i < L_MAX; i += 256) { zr[i] = u0[i]; zi[i] = u1[i]; }
    __syncthreads();

    v2f br[4], bi[4], bin[4];
    make_bfrag(lane, -1.0f, br, bi, bin);              // forward DFT matrix
    fft_stage<0, 1>(zr, zi, lane, wave, br, bi, bin);  __syncthreads();
    fft_stage<1, 2>(zr, zi, lane, wave, br, bi, bin);  __syncthreads();
    fft_stage<2, 0>(zr, zi, lane, wave, br, bi, bin);  __syncthreads();

    const float2* kfh = Kf + (size_t)h * L_MAX;        // already digit-reversed
    for (int i = tid; i < L_MAX; i += 256) {
        float2 k = kfh[i];
        float xr = zr[i], xi = zi[i];
        zr[i] = xr * k.x - xi * k.y;
        zi[i] = xr * k.y + xi * k.x;
    }
    __syncthreads();

    make_bfrag(lane, 1.0f, br, bi, bin);               // conj DFT matrix
    fft_stage<2, 0>(zr, zi, lane, wave, br, bi, bin);  __syncthreads();
    fft_stage<1, 3>(zr, zi, lane, wave, br, bi, bin);  __syncthreads();
    fft_stage<0, 4>(zr, zi, lane, wave, br, bi, bin);  __syncthreads();

    float* y0 = y + ((size_t)(2 * p)     * H_CH + h) * L_MAX;
    float* y1 = y + ((size_t)(2 * p + 1) * H_CH + h) * L_MAX;
    for (int i = tid; i < L_MAX; i += 256) { y0[i] = zr[i]; y1[i] = zi[i]; }
}

// ---------------------------------------------------------------------------
extern "C" void kernel_launch(void* const* d_in, const int* in_sizes, int n_in,
                              void* d_out, int out_size, void* d_ws, size_t ws_size,
                              hipStream_t stream) {
    const float* u         = (const float*)d_in[0];
    const float* Lambda_re = (const float*)d_in[1];
    const float* Lambda_im = (const float*)d_in[2];
    const float* P_re      = (const float*)d_in[3];
    const float* P_im      = (const float*)d_in[4];
    const float* B_re      = (const float*)d_in[5];
    const float* B_im      = (const float*)d_in[6];
    const float* C_re      = (const float*)d_in[7];
    const float* C_im      = (const float*)d_in[8];
    const float* log_step  = (const float*)d_in[9];
    float2* Kf = (float2*)d_ws;                 // 256*4096 complex = 8 MB scratch

    s4_kernelgen<<<dim3(H_CH), dim3(256), 0, stream>>>(
        Lambda_re, Lambda_im, P_re, P_im, B_re, B_im, C_re, C_im, log_step, Kf);

    s4_fftconv<<<dim3(H_CH, BATCH / 2), dim3(256), 0, stream>>>(
        u, Kf, (float*)d_out);
}